// LinearAutoregressiveHMM_90666759619187
// MI455X (gfx1250) — compile-verified
//
#include <hip/hip_runtime.h>
#include <hip/hip_bf16.h>
#include <math.h>

typedef __attribute__((ext_vector_type(2))) float v2f;
typedef __attribute__((ext_vector_type(4))) float v4f;
typedef __attribute__((ext_vector_type(8))) float v8f;

#define NSTATES 8
#define LBW 128
#define CDIM 16
#define BATCH 512
#define PREDW 16
#define LOG2PI 1.8378770664093453f

// One 16x16x4 f32 WMMA taking K-pair P (0 or 1) out of packed v4f fragments.
template <int P>
static __device__ inline v8f do_wmma(v8f acc, v4f a, v4f b) {
  v2f pa = __builtin_shufflevector(a, a, 2 * P, 2 * P + 1);
  v2f pb = __builtin_shufflevector(b, b, 2 * P, 2 * P + 1);
  return __builtin_amdgcn_wmma_f32_16x16x4_f32(false, pa, false, pb, (short)0,
                                               acc, false, false);
}

// ---------------------------------------------------------------------------
// Kernel 0: transpose W into fragment-native layout.
// Wt2[((s*128+l)*2 + half)*16 + d][cc] = W[s][l][8*half+cc][d], cc=0..7.
// Each lane's 8 B-operand floats for a full K=16 step become one 32B run.
// ---------------------------------------------------------------------------
__global__ __launch_bounds__(256)
void transpose_w_kernel(const float* __restrict__ W, float* __restrict__ Wt2) {
  const int o = blockIdx.x * 256 + threadIdx.x;  // 262144 total
  const int cc = o & 7;
  const int d  = (o >> 3) & 15;
  const int hf = (o >> 7) & 1;
  const int l  = (o >> 8) & 127;
  const int s  = o >> 15;
  Wt2[o] = W[((s * LBW + l) * CDIM + 8 * hf + cc) * CDIM + d];
}

// ---------------------------------------------------------------------------
// Kernel 1: per-state precision matrices, logdet, log-softmax of transition /
// initial distribution, and the deterministic argmax next-state map.
// ---------------------------------------------------------------------------
__global__ __launch_bounds__(32)
void prep_kernel(const float* __restrict__ tmat,      // (8,8)
                 const float* __restrict__ initd,     // (8,)
                 const float* __restrict__ covchol,   // (8,16,16)
                 float* __restrict__ prec,            // (8,16,16)
                 float* __restrict__ logdet,          // (8,)
                 float* __restrict__ log_trans,       // (8,8)
                 float* __restrict__ log_pi,          // (8,)
                 int*   __restrict__ next_state) {    // (8,)
  const int t = threadIdx.x;
  if (t < NSTATES) {
    float chol[16][16];
    const float* cc = covchol + t * 256;
    for (int i = 0; i < 16; ++i)
      for (int j = 0; j < 16; ++j) {
        float v = cc[i * 16 + j];
        chol[i][j] = (j < i) ? v : ((j == i) ? expf(v) : 0.f);
      }
    float cov[16][16];
    for (int i = 0; i < 16; ++i)
      for (int j = 0; j <= i; ++j) {
        float s = 0.f;
        for (int k = 0; k <= j; ++k) s += chol[i][k] * chol[j][k];
        if (i == j) s += 1e-6f;
        cov[i][j] = s;
        cov[j][i] = s;
      }
    float Lm[16][16];
    for (int i = 0; i < 16; ++i)
      for (int j = 0; j < 16; ++j) Lm[i][j] = 0.f;
    for (int j = 0; j < 16; ++j) {
      float s = cov[j][j];
      for (int k = 0; k < j; ++k) s -= Lm[j][k] * Lm[j][k];
      float dj = sqrtf(s);
      Lm[j][j] = dj;
      for (int i = j + 1; i < 16; ++i) {
        float v = cov[i][j];
        for (int k = 0; k < j; ++k) v -= Lm[i][k] * Lm[j][k];
        Lm[i][j] = v / dj;
      }
    }
    float ld = 0.f;
    for (int j = 0; j < 16; ++j) ld += logf(Lm[j][j]);
    logdet[t] = 2.f * ld;
    float Li[16][16];
    for (int i = 0; i < 16; ++i)
      for (int j = 0; j < 16; ++j) Li[i][j] = 0.f;
    for (int j = 0; j < 16; ++j) {
      Li[j][j] = 1.f / Lm[j][j];
      for (int i = j + 1; i < 16; ++i) {
        float s = 0.f;
        for (int k = j; k < i; ++k) s += Lm[i][k] * Li[k][j];
        Li[i][j] = -s / Lm[i][i];
      }
    }
    for (int c = 0; c < 16; ++c)
      for (int d = 0; d < 16; ++d) {
        int k0 = (c > d) ? c : d;
        float s = 0.f;
        for (int k = k0; k < 16; ++k) s += Li[k][c] * Li[k][d];
        prec[t * 256 + c * 16 + d] = s;
      }
  } else if (t < 16) {
    const int i = t - 8;
    float m = -1e30f;
    for (int j = 0; j < NSTATES; ++j) m = fmaxf(m, tmat[i * 8 + j]);
    float ssum = 0.f;
    for (int j = 0; j < NSTATES; ++j) ssum += expf(tmat[i * 8 + j] - m);
    int best = 0;
    float bv = tmat[i * 8];
    for (int j = 1; j < NSTATES; ++j)
      if (tmat[i * 8 + j] > bv) { bv = tmat[i * 8 + j]; best = j; }
    next_state[i] = best;
    for (int j = 0; j < NSTATES; ++j)
      log_trans[i * 8 + j] = logf(expf(tmat[i * 8 + j] - m) / ssum + 1e-8f);
  } else if (t == 16) {
    float m = -1e30f;
    for (int j = 0; j < NSTATES; ++j) m = fmaxf(m, initd[j]);
    float ssum = 0.f;
    for (int j = 0; j < NSTATES; ++j) ssum += expf(initd[j] - m);
    float ls = logf(ssum);
    for (int j = 0; j < NSTATES; ++j) log_pi[j] = (initd[j] - m) - ls;
  }
}

// ---------------------------------------------------------------------------
// Kernel 2: WMMA GEMM for ar + quadratic form -> emission log-likelihoods.
// grid = (f=128, btile=16), block = 256 threads = 8 waves, wave w = state s.
// Each wave computes TWO 16(batch)x16(dim) f32 tiles (M=32) so B fragments
// (shared W data) are reused across 8 WMMAs. K permuted so both A and B
// fragments for a K=16 step are contiguous 32B per lane (b128 loads).
// ---------------------------------------------------------------------------
__global__ __launch_bounds__(256)
void emis_kernel(const float* __restrict__ em,     // (512,128,16)
                 const float* __restrict__ Wt2,    // (8,128,2,16,8) transposed
                 const float* __restrict__ means,  // (8,16)
                 const float* __restrict__ prec,   // (8,16,16)
                 const float* __restrict__ logdet, // (8,)
                 float* __restrict__ emis) {       // (512,128,8)
  __shared__ float lds_prec[NSTATES][16][16];
  __shared__ float lds_diff[NSTATES][32][16];
  __shared__ float lds_logdet[NSTATES];
  const int tid = threadIdx.x;
  for (int i = tid; i < NSTATES * 256; i += 256)
    ((float*)lds_prec)[i] = prec[i];
  if (tid < NSTATES) lds_logdet[tid] = logdet[tid];
  __syncthreads();

  const int f    = blockIdx.x;       // output time index
  const int b0   = blockIdx.y * 32;  // batch tile base (two 16-row tiles)
  const int w    = tid >> 5;         // wave id == state s
  const int lane = tid & 31;
  const int hi   = lane >> 4;        // lane half -> owns c = 8*hi .. 8*hi+7
  const int mlo  = lane & 15;        // A row (batch) / B,C col (dim)
  const int s    = w;

  v8f acc0 = {};  // batches b0+0..15
  v8f acc1 = {};  // batches b0+16..31

  // Per-lane base pointers. A: em[b, t, 8*hi + 0..7]. B: Wt2 lane run.
  const float* arow0 = em + (size_t)(b0 + mlo) * (LBW * CDIM) + 8 * hi;
  const float* arow1 = arow0 + 16 * (LBW * CDIM);
  const float* brow  = Wt2 + (size_t)s * (LBW * 2 * CDIM * 8) +
                       (size_t)((LBW - f) * 2 + hi) * (CDIM * 8) + mlo * 8;

  for (int t = 0; t < f; ++t) {
    // A fragments: 2 x b128 per tile (c = 8*hi..8*hi+7 at time t)
    v4f a00 = *(const v4f*)(arow0 + t * CDIM);
    v4f a01 = *(const v4f*)(arow0 + t * CDIM + 4);
    v4f a10 = *(const v4f*)(arow1 + t * CDIM);
    v4f a11 = *(const v4f*)(arow1 + t * CDIM + 4);
    // B fragments: 2 x b128, shared by both tiles (W row l = t + 128 - f)
    v4f bb0 = *(const v4f*)(brow + (size_t)t * (2 * CDIM * 8));
    v4f bb1 = *(const v4f*)(brow + (size_t)t * (2 * CDIM * 8) + 4);

    acc0 = do_wmma<0>(acc0, a00, bb0);
    acc0 = do_wmma<1>(acc0, a00, bb0);
    acc0 = do_wmma<0>(acc0, a01, bb1);
    acc0 = do_wmma<1>(acc0, a01, bb1);
    acc1 = do_wmma<0>(acc1, a10, bb0);
    acc1 = do_wmma<1>(acc1, a10, bb0);
    acc1 = do_wmma<0>(acc1, a11, bb1);
    acc1 = do_wmma<1>(acc1, a11, bb1);
  }

  // diff[m][d] = em[b0+m, f, d] - ar - means[s,d]
  const float mean_d = means[s * 16 + mlo];
#pragma unroll
  for (int j = 0; j < 8; ++j) {
    const int m = j + 8 * hi;
    float e0 = em[(size_t)(b0 + m) * (LBW * CDIM) + f * CDIM + mlo];
    lds_diff[w][m][mlo] = e0 - acc0[j] - mean_d;
    float e1 = em[(size_t)(b0 + 16 + m) * (LBW * CDIM) + f * CDIM + mlo];
    lds_diff[w][16 + m][mlo] = e1 - acc1[j] - mean_d;
  }
  __syncthreads();

  // quad[m] = diff^T * prec * diff ; every lane owns one of the 32 batch rows
  {
    const int m = lane;
    float q = 0.f;
    for (int c = 0; c < 16; ++c) {
      float dc = lds_diff[w][m][c];
      float t2 = 0.f;
      for (int d = 0; d < 16; ++d)
        t2 += lds_prec[s][c][d] * lds_diff[w][m][d];
      q += dc * t2;
    }
    float ll = -0.5f * (q + lds_logdet[s] + (float)CDIM * LOG2PI);
    emis[((size_t)(b0 + m) * LBW + f) * NSTATES + s] = ll;
  }
}

// ---------------------------------------------------------------------------
// Kernel 3: forward logsumexp scan per batch + argmax + next-state chain.
// ---------------------------------------------------------------------------
__global__ __launch_bounds__(256)
void forward_kernel(const float* __restrict__ emis,       // (512,128,8)
                    const float* __restrict__ log_trans,  // (8,8)
                    const float* __restrict__ log_pi,     // (8,)
                    const int*   __restrict__ next_state, // (8,)
                    int* __restrict__ state_seq) {        // (512,16)
  const int b = blockIdx.x * blockDim.x + threadIdx.x;
  if (b >= BATCH) return;
  float lt[64];
#pragma unroll
  for (int i = 0; i < 64; ++i) lt[i] = log_trans[i];
  const float* e = emis + (size_t)b * (LBW * NSTATES);
  float la[NSTATES];
#pragma unroll
  for (int j = 0; j < NSTATES; ++j) la[j] = log_pi[j] + e[j];
  for (int f = 1; f < LBW; ++f) {
    const float* ef = e + f * NSTATES;
    float nla[NSTATES];
#pragma unroll
    for (int j = 0; j < NSTATES; ++j) {
      float m = -1e30f;
#pragma unroll
      for (int i = 0; i < NSTATES; ++i) m = fmaxf(m, la[i] + lt[i * 8 + j]);
      float ssum = 0.f;
#pragma unroll
      for (int i = 0; i < NSTATES; ++i) ssum += expf(la[i] + lt[i * 8 + j] - m);
      nla[j] = m + logf(ssum) + ef[j];
    }
#pragma unroll
    for (int j = 0; j < NSTATES; ++j) la[j] = nla[j];
  }
  int best = 0;
  float bv = la[0];
#pragma unroll
  for (int j = 1; j < NSTATES; ++j)
    if (la[j] > bv) { bv = la[j]; best = j; }
  int st = best;
  for (int k = 0; k < PREDW; ++k) {
    st = next_state[st];
    state_seq[b * PREDW + k] = st;
  }
}

// ---------------------------------------------------------------------------
// Kernel 4: autoregressive rollout, one block per batch.
// ---------------------------------------------------------------------------
__global__ __launch_bounds__(256)
void predict_kernel(const float* __restrict__ em,        // (512,128,16)
                    const float* __restrict__ W,         // (8,128,16,16)
                    const float* __restrict__ means,     // (8,16)
                    const int*   __restrict__ state_seq, // (512,16)
                    float* __restrict__ out) {           // (512,16,16)
  __shared__ float buf[LBW + PREDW][CDIM];
  __shared__ float part[16][16];
  __shared__ int sseq[PREDW];
  const int b = blockIdx.x;
  const int tid = threadIdx.x;
  for (int i = tid; i < LBW * CDIM; i += 256)
    ((float*)buf)[i] = em[(size_t)b * (LBW * CDIM) + i];
  if (tid < PREDW) sseq[tid] = state_seq[b * PREDW + tid];
  __syncthreads();

  const int d = tid & 15;
  const int chunk = tid >> 4;
  for (int k = 0; k < PREDW; ++k) {
    const int s = sseq[k];
    const float* wS = W + (size_t)s * (LBW * CDIM * CDIM);
    float p = 0.f;
    for (int li = 0; li < 8; ++li) {
      const int l = chunk * 8 + li;
      const float* br = buf[k + l];
      const float* wr = wS + l * 256 + d;
#pragma unroll
      for (int c = 0; c < CDIM; ++c) p += br[c] * wr[c * 16];
    }
    part[chunk][d] = p;
    __syncthreads();
    if (tid < 16) {
      float sum = means[s * 16 + tid];
      for (int ch = 0; ch < 16; ++ch) sum += part[ch][tid];
      out[((size_t)b * PREDW + k) * CDIM + tid] = sum;
      buf[LBW + k][tid] = sum;
    }
    __syncthreads();
  }
}

// ---------------------------------------------------------------------------
extern "C" void kernel_launch(void* const* d_in, const int* in_sizes, int n_in,
                              void* d_out, int out_size, void* d_ws, size_t ws_size,
                              hipStream_t stream) {
  const float* emissions = (const float*)d_in[0];  // (512,128,16)
  const float* tmat      = (const float*)d_in[1];  // (8,8)
  const float* initd     = (const float*)d_in[2];  // (8,)
  const float* means     = (const float*)d_in[3];  // (8,16)
  const float* covchol   = (const float*)d_in[4];  // (8,16,16)
  const float* W         = (const float*)d_in[5];  // (8,128,16,16)
  float* out = (float*)d_out;

  char* ws = (char*)d_ws;
  const size_t OFF_EMIS  = 0;                        // 512*128*8 f32 = 2 MB
  const size_t OFF_WT2   = OFF_EMIS + 2097152;       // 262144 f32   = 1 MB
  const size_t OFF_PREC  = OFF_WT2 + 1048576;        // 8*256 f32
  const size_t OFF_LD    = OFF_PREC + 8192;          // 8 f32
  const size_t OFF_LT    = OFF_LD + 32;              // 64 f32
  const size_t OFF_LP    = OFF_LT + 256;             // 8 f32
  const size_t OFF_NS    = OFF_LP + 32;              // 8 i32
  const size_t OFF_SSEQ  = OFF_NS + 32;              // 512*16 i32

  float* ws_emis      = (float*)(ws + OFF_EMIS);
  float* ws_wt2       = (float*)(ws + OFF_WT2);
  float* ws_prec      = (float*)(ws + OFF_PREC);
  float* ws_logdet    = (float*)(ws + OFF_LD);
  float* ws_log_trans = (float*)(ws + OFF_LT);
  float* ws_log_pi    = (float*)(ws + OFF_LP);
  int*   ws_next      = (int*)  (ws + OFF_NS);
  int*   ws_sseq      = (int*)  (ws + OFF_SSEQ);

  transpose_w_kernel<<<1024, 256, 0, stream>>>(W, ws_wt2);
  prep_kernel<<<1, 32, 0, stream>>>(tmat, initd, covchol, ws_prec, ws_logdet,
                                    ws_log_trans, ws_log_pi, ws_next);
  emis_kernel<<<dim3(LBW, BATCH / 32), 256, 0, stream>>>(
      emissions, ws_wt2, means, ws_prec, ws_logdet, ws_emis);
  forward_kernel<<<(BATCH + 255) / 256, 256, 0, stream>>>(
      ws_emis, ws_log_trans, ws_log_pi, ws_next, ws_sseq);
  predict_kernel<<<BATCH, 256, 0, stream>>>(emissions, W, means, ws_sseq, out);
}